// LiftedLoss_86517821215229
// MI455X (gfx1250) — compile-verified
//
#include <hip/hip_runtime.h>
#include <math.h>

typedef float v2f __attribute__((ext_vector_type(2)));
typedef float v8f __attribute__((ext_vector_type(8)));
typedef int   v4i __attribute__((vector_size(16)));

#define MARGIN 0.4f
#define TILE 128
#define TK 32
#define PITCH 36   // 128-row LDS pitch: 16B-aligned float4 stores, conflict-free column reads

#if defined(__gfx1250__) && __has_builtin(__builtin_amdgcn_global_load_async_to_lds_b128)
#define USE_ASYNC 1
#define GPTR(p) ((__attribute__((address_space(1))) v4i*)(p))
#define LPTR(p) ((__attribute__((address_space(3))) v4i*)(p))
#else
#define USE_ASYNC 0
#endif

template <int N>
__device__ __forceinline__ void wait_async() {
#if USE_ASYNC
#if __has_builtin(__builtin_amdgcn_s_wait_asynccnt)
    __builtin_amdgcn_s_wait_asynccnt(N);
#else
    asm volatile("s_wait_asynccnt %0" ::"i"(N) : "memory");
#endif
#endif
}

__global__ void init_ws_kernel(float* wsf, int n) {
    int i = blockIdx.x * blockDim.x + threadIdx.x;
    if (i < n) wsf[i] = 0.0f;
}

__global__ __launch_bounds__(256) void sqnorm_kernel(const float* __restrict__ F,
                                                     float* __restrict__ sq,
                                                     int N, int D) {
    int wave = threadIdx.x >> 5;
    int lane = threadIdx.x & 31;
    int row  = blockIdx.x * 8 + wave;
    if (row >= N) return;
    const float* rp = F + (size_t)row * D;
    float s = 0.0f;
    for (int c = lane * 4; c + 3 < D; c += 128) {
        float4 v = *(const float4*)(rp + c);
        s += v.x * v.x + v.y * v.y + v.z * v.z + v.w * v.w;
    }
    #pragma unroll
    for (int m = 16; m >= 1; m >>= 1) s += __shfl_xor(s, m, 32);
    if (lane == 0) sq[row] = s;
}

// PASS 1: accumulate S[i] = sum_{neg j} exp(margin - d_ij), count positive pairs.
// PASS 2: accumulate sum over positive pairs of max(log(S_i+S_j)+d_ij, 0)^2.
template <int PASS>
__global__ __launch_bounds__(512) void lifted_pass_kernel(
    const float* __restrict__ F, const int* __restrict__ lab,
    const float* __restrict__ sq, float* __restrict__ S,
    float* __restrict__ lossAcc, unsigned* __restrict__ cntAcc,
    int N, int D)
{
    __shared__ __align__(16) float As[2][TILE * PITCH];
    __shared__ __align__(16) float Bs[2][TILE * PITCH];
    __shared__ float s_sq_i[TILE], s_sq_j[TILE];
    __shared__ int   s_lab_i[TILE], s_lab_j[TILE];
    __shared__ float s_row[TILE];
    __shared__ float s_S_i[TILE], s_S_j[TILE];
    __shared__ float s_loss;
    __shared__ unsigned s_cnt;

    const int tid  = threadIdx.x;
    const int lane = tid & 31;
    const int wave = tid >> 5;
    const int wi   = wave >> 2;          // 4x4 wave grid over the 128x128 tile
    const int wj   = wave & 3;
    const int m16  = lane & 15;
    const int kh   = (lane >> 4) << 1;   // A/B fragment K-pair select per half-wave
    const int i0   = blockIdx.y * TILE;
    const int j0   = blockIdx.x * TILE;

    // Per-thread B128 chunk mapping: 1024 chunks per 128x32 tile, 2 per thread per matrix.
    int rows[2], c4s[2];
    #pragma unroll
    for (int p = 0; p < 2; ++p) {
        int idx = tid + p * 512;
        rows[p] = idx >> 3;
        c4s[p]  = (idx & 7) << 2;
    }

    auto issue_slab = [&](int buf, int kk) {
        #pragma unroll
        for (int p = 0; p < 2; ++p) {
            const float* srcA = F + (size_t)(i0 + rows[p]) * D + kk + c4s[p];
            const float* srcB = F + (size_t)(j0 + rows[p]) * D + kk + c4s[p];
            float* dA = &As[buf][rows[p] * PITCH + c4s[p]];
            float* dB = &Bs[buf][rows[p] * PITCH + c4s[p]];
#if USE_ASYNC
            __builtin_amdgcn_global_load_async_to_lds_b128(GPTR(srcA), LPTR(dA), 0, 0);
            __builtin_amdgcn_global_load_async_to_lds_b128(GPTR(srcB), LPTR(dB), 0, 0);
#else
            *(float4*)dA = *(const float4*)srcA;
            *(float4*)dB = *(const float4*)srcB;
#endif
        }
    };

    if (tid < TILE) {
        s_sq_i[tid]  = sq[i0 + tid];
        s_sq_j[tid]  = sq[j0 + tid];
        s_lab_i[tid] = lab[i0 + tid];
        s_lab_j[tid] = lab[j0 + tid];
        s_row[tid]   = 0.0f;
        if (PASS == 2) { s_S_i[tid] = S[i0 + tid]; s_S_j[tid] = S[j0 + tid]; }
    }
    if (tid == 0) { s_loss = 0.0f; s_cnt = 0u; }

    v8f acc[2][2] = {};

    const int T = D / TK;
    issue_slab(0, 0);                         // prologue prefetch (4 async ops per wave)
    for (int t = 0; t < T; ++t) {
        const int cur = t & 1;
        if (t + 1 < T) { issue_slab(cur ^ 1, (t + 1) * TK); wait_async<4>(); }
        else           { wait_async<0>(); }
        __syncthreads();                      // slab t visible to all waves

        const float* Ab = As[cur];
        const float* Bb = Bs[cur];
        #pragma unroll
        for (int kl = 0; kl < TK; kl += 4) {
            v2f a0 = *(const v2f*)&Ab[(wi * 32 +      m16) * PITCH + kl + kh];
            v2f a1 = *(const v2f*)&Ab[(wi * 32 + 16 + m16) * PITCH + kl + kh];
            v2f b0 = *(const v2f*)&Bb[(wj * 32 +      m16) * PITCH + kl + kh];
            v2f b1 = *(const v2f*)&Bb[(wj * 32 + 16 + m16) * PITCH + kl + kh];
            acc[0][0] = __builtin_amdgcn_wmma_f32_16x16x4_f32(false, a0, false, b0, (short)0, acc[0][0], false, false);
            acc[0][1] = __builtin_amdgcn_wmma_f32_16x16x4_f32(false, a0, false, b1, (short)0, acc[0][1], false, false);
            acc[1][0] = __builtin_amdgcn_wmma_f32_16x16x4_f32(false, a1, false, b0, (short)0, acc[1][0], false, false);
            acc[1][1] = __builtin_amdgcn_wmma_f32_16x16x4_f32(false, a1, false, b1, (short)0, acc[1][1], false, false);
        }
        __syncthreads();                      // buffer cur free for prefetch at t+2
    }

    // Epilogue. C/D layout: VGPR v holds row M=v (lanes 0-15) and M=v+8 (lanes 16-31), N = lane%16.
    const int rbase = wi * 32 + ((lane >> 4) << 3);
    unsigned cnt = 0;
    float ls = 0.0f;

    #pragma unroll
    for (int ti = 0; ti < 2; ++ti) {
        #pragma unroll
        for (int v = 0; v < 8; ++v) {
            const int m_loc = rbase + ti * 16 + v;
            const float si  = s_sq_i[m_loc];
            const int   li  = s_lab_i[m_loc];
            float esum = 0.0f;
            #pragma unroll
            for (int tj = 0; tj < 2; ++tj) {
                const int n_loc = wj * 32 + tj * 16 + m16;
                const float g = acc[ti][tj][v];
                const float d = sqrtf(fmaxf(si + s_sq_j[n_loc] - 2.0f * g, 0.0f) + 1e-12f);
                const bool pos = (li == s_lab_j[n_loc]);
                if (PASS == 1) {
                    if (pos) cnt++;
                    else     esum += __expf(MARGIN - d);
                } else {
                    if (pos) {
                        float J = __logf(s_S_i[m_loc] + s_S_j[n_loc]) + d;
                        float r = fmaxf(J, 0.0f);
                        ls += r * r;
                    }
                }
            }
            if (PASS == 1) {
                // reduce across the 16 N-lanes of this half-wave
                esum += __shfl_xor(esum, 1, 16);
                esum += __shfl_xor(esum, 2, 16);
                esum += __shfl_xor(esum, 4, 16);
                esum += __shfl_xor(esum, 8, 16);
                if (m16 == 0) atomicAdd(&s_row[m_loc], esum);
            }
        }
    }

    if (PASS == 1) {
        if (cnt) atomicAdd(&s_cnt, cnt);
        __syncthreads();
        if (tid < TILE) {
            float r = s_row[tid];
            if (r != 0.0f) atomicAdd(&S[i0 + tid], r);
        }
        if (tid == 0 && s_cnt) atomicAdd(cntAcc, s_cnt);
    } else {
        if (ls != 0.0f) atomicAdd(&s_loss, ls);
        __syncthreads();
        if (tid == 0) atomicAdd(lossAcc, s_loss);
    }
}

__global__ void finalize_kernel(const float* __restrict__ lossAcc,
                                const unsigned* __restrict__ cntAcc,
                                float* __restrict__ out) {
    if (threadIdx.x == 0 && blockIdx.x == 0)
        out[0] = lossAcc[0] / (2.0f * (float)cntAcc[0]);
}

extern "C" void kernel_launch(void* const* d_in, const int* in_sizes, int n_in,
                              void* d_out, int out_size, void* d_ws, size_t ws_size,
                              hipStream_t stream) {
    const float* F  = (const float*)d_in[0];
    const int*  lab = (const int*)d_in[1];
    const int N = in_sizes[1];
    const int D = in_sizes[0] / N;

    float*    wsf     = (float*)d_ws;
    float*    sq      = wsf;
    float*    S       = wsf + N;
    float*    lossAcc = wsf + 2 * N;
    unsigned* cntAcc  = (unsigned*)(wsf + 2 * N + 1);

    const int initN = 2 * N + 2;
    init_ws_kernel<<<(initN + 255) / 256, 256, 0, stream>>>(wsf, initN);
    sqnorm_kernel<<<N / 8, 256, 0, stream>>>(F, sq, N, D);

    dim3 grid(N / TILE, N / TILE);
    lifted_pass_kernel<1><<<grid, 512, 0, stream>>>(F, lab, sq, S, lossAcc, cntAcc, N, D);
    lifted_pass_kernel<2><<<grid, 512, 0, stream>>>(F, lab, sq, S, lossAcc, cntAcc, N, D);

    finalize_kernel<<<1, 1, 0, stream>>>(lossAcc, cntAcc, (float*)d_out);
}